// PartitionPadding_70428873720447
// MI455X (gfx1250) — compile-verified
//
#include <hip/hip_runtime.h>
#include <hip/hip_bf16.h>

#define BATCH     2048
#define MAX_ATOMS 640
#define DFEAT     64

// 128-bit vector type that clang's nontemporal builtins accept.
typedef float v4f __attribute__((ext_vector_type(4)));

// ---------------------------------------------------------------------------
// Kernel 1: starts[b] = index of first atom whose molecule id == b
// (== jnp.searchsorted(molecule_indicator, arange(BATCH)) on a sorted array).
// Thread i fills starts[b] for all b in (ind[i-1], ind[i]]; thread 0 fills
// [0, ind[0]]; the last thread fills the tail with N so empty trailing
// molecules get cnt == 0.
// ---------------------------------------------------------------------------
__global__ void pp_compute_starts(const int* __restrict__ ind,
                                  int* __restrict__ starts, int N) {
    int i = blockIdx.x * blockDim.x + threadIdx.x;
    if (i >= N) return;
    int cur = ind[i];
    if (i == 0) {
        for (int b = 0; b <= cur; ++b) starts[b] = 0;
    } else {
        int prev = ind[i - 1];
        for (int b = prev + 1; b <= cur; ++b) starts[b] = i;
    }
    if (i == N - 1) {
        for (int b = cur + 1; b < BATCH; ++b) starts[b] = N;
    }
}

// ---------------------------------------------------------------------------
// Kernel 2: zero ONLY the padding rows [cnt, MAX_ATOMS) of each molecule.
// One block per molecule; 128-bit non-temporal stores (write-once stream,
// keep it out of L2 so the hot index data stays resident).
// ---------------------------------------------------------------------------
__global__ void pp_zero_padding(const int* __restrict__ starts,
                                float* __restrict__ out, int N) {
    int b = blockIdx.x;
    int s = starts[b];
    int e = (b == BATCH - 1) ? N : starts[b + 1];
    int cnt = e - s;
    if (cnt < 0) cnt = 0;
    if (cnt > MAX_ATOMS) cnt = MAX_ATOMS;

    v4f* dst = (v4f*)(out + ((size_t)b * MAX_ATOMS + (size_t)cnt) * DFEAT);
    int n4 = (MAX_ATOMS - cnt) * (DFEAT / 4);
    const v4f z = {0.f, 0.f, 0.f, 0.f};
    for (int t = threadIdx.x; t < n4; t += blockDim.x) {
        __builtin_nontemporal_store(z, dst + t);
    }
}

// ---------------------------------------------------------------------------
// Kernel 3: scatter. 16 lanes cooperate per atom row (64 floats = 16 x
// 128-bit): each lane does one 128-bit NT load + one 128-bit NT store, so a
// wave32 moves two fully-coalesced 256 B rows. Index/starts loads stay on the
// cached path (shared across lanes). pos = atom - starts[mol] reproduces the
// reference's intra-molecule offset; pos >= MAX_ATOMS is dropped (JAX scatter
// out-of-bounds drop semantics).
// ---------------------------------------------------------------------------
__global__ void pp_scatter(const float* __restrict__ feat,
                           const int* __restrict__ ind,
                           const int* __restrict__ starts,
                           float* __restrict__ out, int N) {
    int gid  = blockIdx.x * blockDim.x + threadIdx.x;
    int atom = gid >> 4;     // 16 lanes per atom
    int comp = gid & 15;     // which 16B chunk of the 64-float row
    if (atom >= N) return;

    int mol = ind[atom];
    int pos = atom - starts[mol];
    if (pos < 0 || pos >= MAX_ATOMS) return;

    const v4f* src = (const v4f*)(feat + (size_t)atom * DFEAT);
    v4f*       dst = (v4f*)(out + ((size_t)mol * MAX_ATOMS + (size_t)pos) * DFEAT);
    v4f v = __builtin_nontemporal_load(src + comp);
    __builtin_nontemporal_store(v, dst + comp);
}

extern "C" void kernel_launch(void* const* d_in, const int* in_sizes, int n_in,
                              void* d_out, int out_size, void* d_ws, size_t ws_size,
                              hipStream_t stream) {
    const float* atom_features      = (const float*)d_in[0];
    const int*   molecule_indicator = (const int*)d_in[1];
    float*       out               = (float*)d_out;
    int*         starts            = (int*)d_ws;   // BATCH ints of scratch

    const int N = in_sizes[1];                      // 1,000,000 atoms

    // 1) segment starts (searchsorted on the sorted indicator)
    {
        int threads = 256;
        int blocks  = (N + threads - 1) / threads;
        pp_compute_starts<<<blocks, threads, 0, stream>>>(molecule_indicator, starts, N);
    }

    // 2) zero only the padding region of the output
    {
        pp_zero_padding<<<BATCH, 256, 0, stream>>>(starts, out, N);
    }

    // 3) scatter atom rows with 128-bit non-temporal vector memory ops
    {
        int threads     = 256;
        long long work  = (long long)N * 16;        // 16 lanes per atom
        int blocks      = (int)((work + threads - 1) / threads);
        pp_scatter<<<blocks, threads, 0, stream>>>(atom_features, molecule_indicator,
                                                   starts, out, N);
    }
}